// SAGEBC_49486613185053
// MI455X (gfx1250) — compile-verified
//
#include <hip/hip_runtime.h>
#include <hip/hip_bf16.h>

typedef __attribute__((ext_vector_type(2))) float v2f;
typedef __attribute__((ext_vector_type(8))) float v8f;

#define DFEAT 128

// ---------------------------------------------------------------- utilities
__global__ void zero_f32(float* __restrict__ p, long long n) {
  long long i = (long long)blockIdx.x * blockDim.x + threadIdx.x;
  if (i < n) p[i] = 0.0f;
}

__global__ void deg_count(const int* __restrict__ dst, float* __restrict__ deg, int E) {
  int i = blockIdx.x * blockDim.x + threadIdx.x;
  if (i < E) atomicAdd(&deg[dst[i]], 1.0f);
}

__global__ void inv_deg_kernel(float* __restrict__ deg, int n) {
  int i = blockIdx.x * blockDim.x + threadIdx.x;
  if (i < n) deg[i] = 1.0f / fmaxf(deg[i], 1.0f);
}

// ------------------------------------------------------------ weight repack
// Interleave W[K x NOUT] (row-major) so that each lane's WMMA B operand
// (rows k,k+1 of one column) is one contiguous float2 -> global_load_b64:
//   Wp[ (k>>2)*NOUT*4 + ((k&3)>>1)*NOUT*2 + col*2 + (k&1) ] = W[k*NOUT+col]
__global__ void repack_w(const float* __restrict__ W, float* __restrict__ Wp,
                         int K, int NOUT) {
  int i = blockIdx.x * blockDim.x + threadIdx.x;
  if (i >= K * NOUT) return;
  int k = i / NOUT, col = i - k * NOUT;
  Wp[(k >> 2) * NOUT * 4 + ((k & 3) >> 1) * NOUT * 2 + col * 2 + (k & 1)] = W[i];
}

// ------------------------------------------------------------- dense GEMM
// out[i,:] = (RELU_IN ? relu(h[i,:]) : h[i,:]) @ W  (+ bias if BIAS)
// Block = 256 threads = 8 waves; wave owns a 16-row strip x all NOUT cols.
// K = 128 stepped by 4 with V_WMMA_F32_16X16X4_F32.
//
// fp32 WMMA operand layout (ISA 7.12.2):
//   A 16x4 : lane&15 = row M; lanes<16 hold K={0,1} in {.x,.y}, lanes>=16 K={2,3}
//   B 4x16 : lane&15 = col N; lanes<16 hold K={0,1}, lanes>=16 K={2,3}
//   C 16x16: VGPR r -> M=r (lanes<16) / M=r+8 (lanes>=16), N = lane&15
template <int NOUT, bool BIAS, bool RELU_IN>
__global__ __launch_bounds__(256) void gemm_wmma(
    const float* __restrict__ h, const float* __restrict__ Wp,
    const float* __restrict__ bias, float* __restrict__ out, int nrows) {
  constexpr int NT = NOUT / 16;
  const int lane = threadIdx.x & 31;
  const int wave = threadIdx.x >> 5;
  const int m = lane & 15;
  const int rowbase = blockIdx.x * 128 + wave * 16;

  // Clamp rows for loads (EXEC stays all-ones through the WMMAs, per the ISA
  // restriction); stores are guarded instead.
  int row = rowbase + m;
  int rowc = row < nrows ? row : (nrows - 1);
  const int ksel = (lane < 16) ? 0 : 2;

  const float* __restrict__ hrow = h + (size_t)rowc * DFEAT;
  const float* __restrict__ wb = Wp + (lane < 16 ? 0 : NOUT * 2) + m * 2;

  v8f acc[NT] = {};

  for (int kb = 0; kb < 32; ++kb) {
    const int k = kb * 4 + ksel;
    v2f a;
    a.x = hrow[k];
    a.y = hrow[k + 1];
    if (RELU_IN) {
      a.x = fmaxf(a.x, 0.0f);
      a.y = fmaxf(a.y, 0.0f);
    }
    const float* __restrict__ wk = wb + kb * NOUT * 4;
#pragma unroll
    for (int t = 0; t < NT; ++t) {
      v2f b = *(const v2f*)(wk + t * 32);  // cols t*16+m, rows k..k+1 (b64)
      acc[t] = __builtin_amdgcn_wmma_f32_16x16x4_f32(
          false, a, false, b, (short)0, acc[t], false, false);
    }
  }

  const int rhalf = (lane < 16) ? 0 : 8;
#pragma unroll
  for (int t = 0; t < NT; ++t) {
    const int col = t * 16 + m;
    const float bv = BIAS ? bias[col] : 0.0f;
#pragma unroll
    for (int r = 0; r < 8; ++r) {
      const int rr = rowbase + rhalf + r;
      if (rr < nrows) out[(size_t)rr * NOUT + col] = acc[t][r] + bv;
    }
  }
}

// ----------------------------------------------- scaled scatter-accumulate
// out[dst,:] += p[src,:] * inv_deg[dst]   (one wave32 per edge, coalesced
// gather + coalesced global_atomic_add_f32; aggregation done AFTER the
// neighbor GEMM, so layer 2 only moves 64 columns)
template <int DCOL>
__global__ __launch_bounds__(256) void agg_scaled(
    const float* __restrict__ p, const int* __restrict__ src,
    const int* __restrict__ dst, const float* __restrict__ invdeg,
    float* __restrict__ out, int E) {
  int t = blockIdx.x * blockDim.x + threadIdx.x;
  int e = t >> 5;
  int lane = t & 31;
  if (e >= E) return;
  int s = src[e];
  int d = dst[e];
  float w = invdeg[d];
  constexpr int V = DCOL / 32;  // floats per lane: 4 (DCOL=128) or 2 (DCOL=64)
  const float* sp = p + (size_t)s * DCOL + lane * V;
  float* ap = out + (size_t)d * DCOL + lane * V;
  if (V == 4) {
    float4 v = *(const float4*)sp;
    atomicAdd(ap + 0, v.x * w);
    atomicAdd(ap + 1, v.y * w);
    atomicAdd(ap + 2, v.z * w);
    atomicAdd(ap + 3, v.w * w);
  } else {
    float2 v = *(const float2*)sp;
    atomicAdd(ap + 0, v.x * w);
    atomicAdd(ap + 1, v.y * w);
  }
}

// ------------------------------------------------------------------ driver
extern "C" void kernel_launch(void* const* d_in, const int* in_sizes, int n_in,
                              void* d_out, int out_size, void* d_ws, size_t ws_size,
                              hipStream_t stream) {
  const float* x   = (const float*)d_in[0];
  const int*   src = (const int*)d_in[1];
  const int*   dst = (const int*)d_in[2];
  const float* Ws0 = (const float*)d_in[3];
  const float* Wn0 = (const float*)d_in[4];
  const float* b0  = (const float*)d_in[5];
  const float* Ws1 = (const float*)d_in[6];
  const float* Wn1 = (const float*)d_in[7];
  const float* b1  = (const float*)d_in[8];
  const float* Ws2 = (const float*)d_in[9];
  const float* Wn2 = (const float*)d_in[10];
  const float* b2  = (const float*)d_in[11];

  const int N = in_sizes[0] / DFEAT;
  const int E = in_sizes[1];

  // workspace: [inv_deg Npad][6 packed weights][bufA N*128][bufB N*128]
  float* deg = (float*)d_ws;
  float* Ps0 = deg + (((size_t)N + 127) & ~(size_t)127);
  float* Pn0 = Ps0 + 128 * 128;
  float* Ps1 = Pn0 + 128 * 128;
  float* Pn1 = Ps1 + 128 * 128;
  float* Ps2 = Pn1 + 128 * 128;
  float* Pn2 = Ps2 + 128 * 64;
  float* bufA = Pn2 + 128 * 64;
  float* bufB = bufA + (size_t)N * DFEAT;

  const int TB = 256;
  auto cdiv = [](long long a, long long b) { return (int)((a + b - 1) / b); };
  const int aggBlocks  = cdiv((long long)E * 32, TB);
  const int gemmBlocks = cdiv(N, 128);
  const int rpBlocksH  = cdiv(128 * 128, TB);
  const int rpBlocksO  = cdiv(128 * 64, TB);

  // degree -> 1/max(deg,1), shared by all layers
  zero_f32<<<cdiv(N, TB), TB, 0, stream>>>(deg, N);
  deg_count<<<cdiv(E, TB), TB, 0, stream>>>(dst, deg, E);
  inv_deg_kernel<<<cdiv(N, TB), TB, 0, stream>>>(deg, N);

  // repack weights into WMMA B-operand-friendly interleave (tiny, 0.6 MB)
  repack_w<<<rpBlocksH, TB, 0, stream>>>(Ws0, Ps0, 128, 128);
  repack_w<<<rpBlocksH, TB, 0, stream>>>(Wn0, Pn0, 128, 128);
  repack_w<<<rpBlocksH, TB, 0, stream>>>(Ws1, Ps1, 128, 128);
  repack_w<<<rpBlocksH, TB, 0, stream>>>(Wn1, Pn1, 128, 128);
  repack_w<<<rpBlocksO, TB, 0, stream>>>(Ws2, Ps2, 128, 64);
  repack_w<<<rpBlocksO, TB, 0, stream>>>(Wn2, Pn2, 128, 64);

  // ---- layer 0: h = x -> bufB (pre-relu; relu folded into layer-1 A loads)
  gemm_wmma<128, true, false><<<gemmBlocks, TB, 0, stream>>>(x, Ps0, b0, bufB, N);
  gemm_wmma<128, false, false><<<gemmBlocks, TB, 0, stream>>>(x, Pn0, nullptr, bufA, N);
  agg_scaled<128><<<aggBlocks, TB, 0, stream>>>(bufA, src, dst, deg, bufB, E);

  // ---- layer 1: h = relu(bufB) -> bufA (neighbor GEMM in place over bufB:
  // each wave reads its own 16 rows before writing them)
  gemm_wmma<128, true, true><<<gemmBlocks, TB, 0, stream>>>(bufB, Ps1, b1, bufA, N);
  gemm_wmma<128, false, true><<<gemmBlocks, TB, 0, stream>>>(bufB, Pn1, nullptr, bufB, N);
  agg_scaled<128><<<aggBlocks, TB, 0, stream>>>(bufB, src, dst, deg, bufA, E);

  // ---- layer 2: h = relu(bufA) -> d_out (64 cols, no final relu)
  gemm_wmma<64, true, true><<<gemmBlocks, TB, 0, stream>>>(bufA, Ps2, b2, (float*)d_out, N);
  gemm_wmma<64, false, true><<<gemmBlocks, TB, 0, stream>>>(bufA, Pn2, nullptr, bufB, N);
  agg_scaled<64><<<aggBlocks, TB, 0, stream>>>(bufB, src, dst, deg, (float*)d_out, E);
}